// HyperbolicPhasorBlock_37082747633869
// MI455X (gfx1250) — compile-verified
//
#include <hip/hip_runtime.h>
#include <hip/hip_bf16.h>
#include <math.h>

typedef float v2f __attribute__((ext_vector_type(2)));
typedef float v8f __attribute__((ext_vector_type(8)));

// D = A(16x4,f32) x B(4x16,f32) + C(16x16,f32)
#define WMMA_F32(a, b, c) \
    __builtin_amdgcn_wmma_f32_16x16x4_f32(false, (a), false, (b), (short)0, (c), false, false)

__device__ __forceinline__ float gelu_exact(float x) {
    return 0.5f * x * (1.0f + erff(x * 0.70710678118654752440f));
}

// Async copy 16B global -> LDS (CDNA5 GLOBAL_LOAD_ASYNC_TO_LDS_B128, ASYNCcnt).
// lds_off = low 32 bits of the generic shared-memory pointer (LDS aperture
// truncation, ISA 10.2).
__device__ __forceinline__ void async_copy_b128(const float* g, const float* lds_ptr) {
    unsigned lds_off = (unsigned)(size_t)lds_ptr;
    unsigned long long ga = (unsigned long long)(size_t)g;
    asm volatile("global_load_async_to_lds_b128 %0, %1, off"
                 :: "v"(lds_off), "v"(ga) : "memory");
}
__device__ __forceinline__ void wait_async0() {
    asm volatile("s_wait_asynccnt 0x0" ::: "memory");
}

// ---------------------------------------------------------------------------
// fp32 WMMA GEMM: C = act(A@B + bias) [+ resid]
// A: MxK row-major, B: KxN row-major, C/resid: MxN row-major.
// Block = 256 threads (8 waves) computes a 128x64 C tile; each wave owns a
// 2x2 grid of 16x16 accumulators. Double-buffered LDS, async loads to LDS.
// A tile LDS stride = 20 floats (80B): 16B-aligned rows for b128 async writes
// AND bank-conflict-free fragment reads (20*r mod 64 distinct, r=0..15).
// ---------------------------------------------------------------------------
__global__ __launch_bounds__(256) void gemm_wmma_f32(
    const float* __restrict__ A, const float* __restrict__ B,
    const float* __restrict__ bias, const float* __restrict__ resid,
    float* __restrict__ C, int M, int N, int Kd, int act)
{
    __shared__ __align__(16) float As[2][128 * 20];   // [buf][row][k], stride 20
    __shared__ __align__(16) float Bs[2][16 * 64];    // [buf][k][n]

    const int tid  = threadIdx.x;
    const int wave = tid >> 5;
    const int lane = tid & 31;
    const int half = lane >> 4;
    const int lr   = lane & 15;

    const int m0 = blockIdx.y * 128;
    const int n0 = blockIdx.x * 64;

    const int colBase = (wave & 1) * 32;   // cols {colBase, colBase+16}
    const int rowBase = (wave >> 1) * 32;  // rows {rowBase, rowBase+16}

    v8f acc00 = {}; v8f acc01 = {}; v8f acc10 = {}; v8f acc11 = {};

    // Loaders: A tile 128x16 (256 thr x 2 float4), B tile 16x64 (256 thr x 1).
    const int arow = tid >> 1, aq = (tid & 1) * 8;
    const int brow = tid >> 4, bcol = (tid & 15) * 4;

    const float* gA = A + (size_t)(m0 + arow) * Kd + aq;
    const float* gB = B + (size_t)brow * N + n0 + bcol;

    // prologue: stage k-block 0 into buffer 0
    async_copy_b128(gA,     &As[0][arow * 20 + aq]);
    async_copy_b128(gA + 4, &As[0][arow * 20 + aq + 4]);
    async_copy_b128(gB,     &Bs[0][brow * 64 + bcol]);

    int buf = 0;
    for (int kb = 0; kb < Kd; kb += 16) {
        wait_async0();          // this wave's copies into `buf` are done
        __syncthreads();        // everyone's copies are done; prev reads done

        if (kb + 16 < Kd) {     // overlap next tile copy with compute
            const float* nA = gA + kb + 16;
            const float* nB = gB + (size_t)(kb + 16) * N;
            async_copy_b128(nA,     &As[buf ^ 1][arow * 20 + aq]);
            async_copy_b128(nA + 4, &As[buf ^ 1][arow * 20 + aq + 4]);
            async_copy_b128(nB,     &Bs[buf ^ 1][brow * 64 + bcol]);
        }

        const float* Asb = As[buf];
        const float* Bsb = Bs[buf];
#pragma unroll
        for (int k0 = 0; k0 < 16; k0 += 4) {
            v2f b0, b1, a0, a1;
            b0.x = Bsb[(k0 + 0 + 2 * half) * 64 + colBase + lr];
            b0.y = Bsb[(k0 + 1 + 2 * half) * 64 + colBase + lr];
            b1.x = Bsb[(k0 + 0 + 2 * half) * 64 + colBase + 16 + lr];
            b1.y = Bsb[(k0 + 1 + 2 * half) * 64 + colBase + 16 + lr];
            a0.x = Asb[(rowBase + lr) * 20 + k0 + 0 + 2 * half];
            a0.y = Asb[(rowBase + lr) * 20 + k0 + 1 + 2 * half];
            a1.x = Asb[(rowBase + 16 + lr) * 20 + k0 + 0 + 2 * half];
            a1.y = Asb[(rowBase + 16 + lr) * 20 + k0 + 1 + 2 * half];
            acc00 = WMMA_F32(a0, b0, acc00);
            acc01 = WMMA_F32(a0, b1, acc01);
            acc10 = WMMA_F32(a1, b0, acc10);
            acc11 = WMMA_F32(a1, b1, acc11);
        }
        buf ^= 1;
    }

    const int nA0 = n0 + colBase + lr;
    const int nA1 = nA0 + 16;
    const float bz0 = bias ? bias[nA0] : 0.0f;
    const float bz1 = bias ? bias[nA1] : 0.0f;
#pragma unroll
    for (int v = 0; v < 8; ++v) {
        const int mA = m0 + rowBase + v + 8 * half;
        const int mB = mA + 16;
        float r00 = acc00[v] + bz0, r01 = acc01[v] + bz1;
        float r10 = acc10[v] + bz0, r11 = acc11[v] + bz1;
        if (act == 1) {
            r00 = gelu_exact(r00); r01 = gelu_exact(r01);
            r10 = gelu_exact(r10); r11 = gelu_exact(r11);
        }
        if (resid) {
            r00 += resid[(size_t)mA * N + nA0]; r01 += resid[(size_t)mA * N + nA1];
            r10 += resid[(size_t)mB * N + nA0]; r11 += resid[(size_t)mB * N + nA1];
        }
        C[(size_t)mA * N + nA0] = r00;
        C[(size_t)mA * N + nA1] = r01;
        C[(size_t)mB * N + nA0] = r10;
        C[(size_t)mB * N + nA1] = r11;
    }
}

// ---------------------------------------------------------------------------
// Elementwise phasor: raw (tok, 2K interleaved pairs) -> [real | imag].
// ---------------------------------------------------------------------------
__device__ __forceinline__ void phasor2(float x0, float x1, float* re, float* im) {
    float px = tanhf(x0) * 0.9f;
    float py = tanhf(x1) * 0.9f;
    float n  = sqrtf(px * px + py * py);
    float nm = fmaxf(n, 1e-5f);
    float sc = fminf((1.0f - 1e-5f) / nm, 1.0f);
    px *= sc; py *= sc;
    float r = n * sc;
    float ph = atan2f(py, px);
    float amp = 1.0f - r;
    *re = amp * cosf(ph);
    *im = amp * sinf(ph);
}

__global__ void phasor_kernel(const float* __restrict__ Kraw,
                              const float* __restrict__ Qraw,
                              float* __restrict__ K2, float* __restrict__ Q2, int n)
{
    int i = blockIdx.x * blockDim.x + threadIdx.x;
    if (i >= n) return;
    int tok = i >> 6, k = i & 63;
    size_t base = (size_t)tok * 128;
    float re, im;
    phasor2(Kraw[base + 2 * k], Kraw[base + 2 * k + 1], &re, &im);
    K2[base + k] = re; K2[base + 64 + k] = im;
    phasor2(Qraw[base + 2 * k], Qraw[base + 2 * k + 1], &re, &im);
    Q2[base + k] = re; Q2[base + 64 + k] = im;
}

// ---------------------------------------------------------------------------
// Per-chunk masked scores: S_c = tril(Q2_c @ K2_c^T), chunk = 64 tokens.
// ---------------------------------------------------------------------------
__global__ __launch_bounds__(256) void s_chunk_kernel(
    const float* __restrict__ Q2, const float* __restrict__ K2,
    float* __restrict__ Sbuf)
{
    const int bc = blockIdx.x;            // b*16 + c
    const size_t tok0 = (size_t)(bc >> 4) * 1024 + (size_t)(bc & 15) * 64;
    const int tid  = threadIdx.x;
    const int wave = tid >> 5, lane = tid & 31, half = lane >> 4, lr = lane & 15;
    const int nOff = (wave & 3) * 16, rOff = (wave >> 2) * 32;

    const float* Qb = Q2 + tok0 * 128;
    const float* Kb = K2 + tok0 * 128;

    v8f acc0 = {}; v8f acc1 = {};
#pragma unroll 8
    for (int k0 = 0; k0 < 128; k0 += 4) {
        v2f bf;   // B[k][n] = K2[tok0+n][k]  (transposed read)
        bf.x = Kb[(size_t)(nOff + lr) * 128 + k0 + 0 + 2 * half];
        bf.y = Kb[(size_t)(nOff + lr) * 128 + k0 + 1 + 2 * half];
        v2f a0;
        a0.x = Qb[(size_t)(rOff + lr) * 128 + k0 + 0 + 2 * half];
        a0.y = Qb[(size_t)(rOff + lr) * 128 + k0 + 1 + 2 * half];
        v2f a1;
        a1.x = Qb[(size_t)(rOff + 16 + lr) * 128 + k0 + 0 + 2 * half];
        a1.y = Qb[(size_t)(rOff + 16 + lr) * 128 + k0 + 1 + 2 * half];
        acc0 = WMMA_F32(a0, bf, acc0);
        acc1 = WMMA_F32(a1, bf, acc1);
    }
    float* Sb = Sbuf + (size_t)bc * 4096;
    const int n = nOff + lr;
#pragma unroll
    for (int v = 0; v < 8; ++v) {
        int mA = rOff + v + 8 * half, mB = mA + 16;
        Sb[mA * 64 + n] = (mA >= n) ? acc0[v] : 0.0f;
        Sb[mB * 64 + n] = (mB >= n) ? acc1[v] : 0.0f;
    }
}

// ---------------------------------------------------------------------------
// Chunked causal linear-attention scan.
// Grid = B * (512/64) = 16 blocks; each owns a 64-wide d-slice and keeps the
// running 128x64 state M = sum K2^T V in LDS. Per chunk:
//   out_c = S_c @ V_c + Q2_c @ M ;  M += K2_c^T @ V_c
// ---------------------------------------------------------------------------
__global__ __launch_bounds__(256) void attn_scan_kernel(
    const float* __restrict__ Q2, const float* __restrict__ K2,
    const float* __restrict__ V,  const float* __restrict__ Sbuf,
    float* __restrict__ R)
{
    __shared__ float Msh[128 * 64];   // 32 KB state
    const int bd = blockIdx.x;        // b*8 + dtile
    const int b  = bd >> 3;
    const int n0 = (bd & 7) * 64;

    const int tid  = threadIdx.x;
    const int wave = tid >> 5, lane = tid & 31, half = lane >> 4, lr = lane & 15;
    const int nOff = (wave & 3) * 16, rOff = (wave >> 2) * 32;

    for (int i = tid; i < 128 * 64; i += 256) Msh[i] = 0.0f;
    __syncthreads();

    for (int c = 0; c < 16; ++c) {
        const size_t tok0 = (size_t)b * 1024 + (size_t)c * 64;
        const float* Sb = Sbuf + (size_t)(b * 16 + c) * 4096;
        const float* Vb = V  + tok0 * 512 + n0;
        const float* Qb = Q2 + tok0 * 128;
        const float* Kb = K2 + tok0 * 128;

        // ---- phase 1: out_c = S_c @ V_c + Q2_c @ M (reads M) ----
        v8f acc0 = {}; v8f acc1 = {};
#pragma unroll 4
        for (int k0 = 0; k0 < 64; k0 += 4) {
            v2f bf;
            bf.x = Vb[(size_t)(k0 + 0 + 2 * half) * 512 + nOff + lr];
            bf.y = Vb[(size_t)(k0 + 1 + 2 * half) * 512 + nOff + lr];
            v2f a0;
            a0.x = Sb[(rOff + lr) * 64 + k0 + 0 + 2 * half];
            a0.y = Sb[(rOff + lr) * 64 + k0 + 1 + 2 * half];
            v2f a1;
            a1.x = Sb[(rOff + 16 + lr) * 64 + k0 + 0 + 2 * half];
            a1.y = Sb[(rOff + 16 + lr) * 64 + k0 + 1 + 2 * half];
            acc0 = WMMA_F32(a0, bf, acc0);
            acc1 = WMMA_F32(a1, bf, acc1);
        }
#pragma unroll 4
        for (int k0 = 0; k0 < 128; k0 += 4) {
            v2f bf;
            bf.x = Msh[(k0 + 0 + 2 * half) * 64 + nOff + lr];
            bf.y = Msh[(k0 + 1 + 2 * half) * 64 + nOff + lr];
            v2f a0;
            a0.x = Qb[(size_t)(rOff + lr) * 128 + k0 + 0 + 2 * half];
            a0.y = Qb[(size_t)(rOff + lr) * 128 + k0 + 1 + 2 * half];
            v2f a1;
            a1.x = Qb[(size_t)(rOff + 16 + lr) * 128 + k0 + 0 + 2 * half];
            a1.y = Qb[(size_t)(rOff + 16 + lr) * 128 + k0 + 1 + 2 * half];
            acc0 = WMMA_F32(a0, bf, acc0);
            acc1 = WMMA_F32(a1, bf, acc1);
        }
        {
            float* Rb = R + tok0 * 512 + n0;
            const int n = nOff + lr;
#pragma unroll
            for (int v = 0; v < 8; ++v) {
                Rb[(size_t)(rOff + v + 8 * half) * 512 + n]      = acc0[v];
                Rb[(size_t)(rOff + 16 + v + 8 * half) * 512 + n] = acc1[v];
            }
        }
        __syncthreads();

        // ---- phase 2: M += K2_c^T @ V_c (each wave owns 4 disjoint subtiles) ----
#pragma unroll
        for (int j = 0; j < 4; ++j) {
            const int mrow0 = ((wave >> 2) * 4 + j) * 16;   // 0..112
            v8f acc = {};
#pragma unroll
            for (int v = 0; v < 8; ++v)
                acc[v] = Msh[(mrow0 + v + 8 * half) * 64 + nOff + lr];
#pragma unroll 4
            for (int k0 = 0; k0 < 64; k0 += 4) {
                v2f bf;
                bf.x = Vb[(size_t)(k0 + 0 + 2 * half) * 512 + nOff + lr];
                bf.y = Vb[(size_t)(k0 + 1 + 2 * half) * 512 + nOff + lr];
                v2f af;   // A[m][l] = K2[tok0+l][m]  (transposed read)
                af.x = Kb[(size_t)(k0 + 0 + 2 * half) * 128 + mrow0 + lr];
                af.y = Kb[(size_t)(k0 + 1 + 2 * half) * 128 + mrow0 + lr];
                acc = WMMA_F32(af, bf, acc);
            }
#pragma unroll
            for (int v = 0; v < 8; ++v)
                Msh[(mrow0 + v + 8 * half) * 64 + nOff + lr] = acc[v];
        }
        __syncthreads();
    }
}

// ---------------------------------------------------------------------------
// Position-normalize + LayerNorm. One wave32 per 512-wide row; in-place safe
// (each lane reads its 16 elements into registers before any write).
// ---------------------------------------------------------------------------
__global__ __launch_bounds__(256) void ln_kernel(
    const float* __restrict__ R, const float* __restrict__ g,
    const float* __restrict__ bta, float* __restrict__ RN)
{
    const int row  = blockIdx.x * 8 + (threadIdx.x >> 5);
    const int lane = threadIdx.x & 31;
    const float* src = R + (size_t)row * 512;
    const int l = row & 1023;
    const float inorm = rsqrtf((float)(l + 1) * 64.0f);

    float vals[16];
    float s = 0.0f, s2 = 0.0f;
#pragma unroll
    for (int j = 0; j < 16; ++j) {
        float v = src[lane + 32 * j] * inorm;
        vals[j] = v; s += v; s2 += v * v;
    }
#pragma unroll
    for (int o = 16; o >= 1; o >>= 1) {
        s  += __shfl_xor(s,  o, 32);
        s2 += __shfl_xor(s2, o, 32);
    }
    const float mu  = s * (1.0f / 512.0f);
    const float var = s2 * (1.0f / 512.0f) - mu * mu;
    const float inv = rsqrtf(var + 1e-5f);
    float* dst = RN + (size_t)row * 512;
#pragma unroll
    for (int j = 0; j < 16; ++j) {
        int e = lane + 32 * j;
        dst[e] = (vals[j] - mu) * inv * g[e] + bta[e];
    }
}

// ---------------------------------------------------------------------------
extern "C" void kernel_launch(void* const* d_in, const int* in_sizes, int n_in,
                              void* d_out, int out_size, void* d_ws, size_t ws_size,
                              hipStream_t stream) {
    const float* x   = (const float*)d_in[0];
    const float* Wk1 = (const float*)d_in[1];
    const float* bk1 = (const float*)d_in[2];
    const float* Wk2 = (const float*)d_in[3];
    const float* bk2 = (const float*)d_in[4];
    const float* Wq1 = (const float*)d_in[5];
    const float* bq1 = (const float*)d_in[6];
    const float* Wq2 = (const float*)d_in[7];
    const float* bq2 = (const float*)d_in[8];
    const float* Wv  = (const float*)d_in[9];
    const float* bv  = (const float*)d_in[10];
    const float* lng = (const float*)d_in[11];
    const float* lnb = (const float*)d_in[12];
    const float* Wo  = (const float*)d_in[13];
    const float* bo  = (const float*)d_in[14];
    float* out = (float*)d_out;

    const int M = 2048;      // B*L tokens
    const int D = 512;
    const int N2K = 128;     // 2*K_OSC

    float* ws   = (float*)d_ws;
    float* Hk   = ws;
    float* Hq   = Hk   + (size_t)M * D;
    float* Vm   = Hq   + (size_t)M * D;
    float* Kraw = Vm   + (size_t)M * D;
    float* Qraw = Kraw + (size_t)M * N2K;
    float* K2   = Qraw + (size_t)M * N2K;
    float* Q2   = K2   + (size_t)M * N2K;
    float* Sbuf = Q2   + (size_t)M * N2K;
    float* R    = Sbuf + (size_t)32 * 4096;

    dim3 blk(256);

    // MLP first layers (gelu) + V projection
    gemm_wmma_f32<<<dim3(D / 64, M / 128), blk, 0, stream>>>(x, Wk1, bk1, nullptr, Hk, M, D, D, 1);
    gemm_wmma_f32<<<dim3(D / 64, M / 128), blk, 0, stream>>>(x, Wq1, bq1, nullptr, Hq, M, D, D, 1);
    gemm_wmma_f32<<<dim3(D / 64, M / 128), blk, 0, stream>>>(x, Wv,  bv,  nullptr, Vm, M, D, D, 0);
    // MLP second layers
    gemm_wmma_f32<<<dim3(N2K / 64, M / 128), blk, 0, stream>>>(Hk, Wk2, bk2, nullptr, Kraw, M, N2K, D, 0);
    gemm_wmma_f32<<<dim3(N2K / 64, M / 128), blk, 0, stream>>>(Hq, Wq2, bq2, nullptr, Qraw, M, N2K, D, 0);
    // Phasor features
    phasor_kernel<<<(M * 64 + 255) / 256, blk, 0, stream>>>(Kraw, Qraw, K2, Q2, M * 64);
    // Per-chunk masked scores
    s_chunk_kernel<<<32, blk, 0, stream>>>(Q2, K2, Sbuf);
    // Chunked causal linear-attention scan
    attn_scan_kernel<<<16, blk, 0, stream>>>(Q2, K2, Vm, Sbuf, R);
    // Position-normalize + LayerNorm (in-place on R)
    ln_kernel<<<M / 8, blk, 0, stream>>>(R, lng, lnb, R);
    // Output projection with residual: out = x + LN(r) @ Wo + bo
    gemm_wmma_f32<<<dim3(D / 64, M / 128), blk, 0, stream>>>(R, Wo, bo, x, out, M, D, D, 0);
}